// VectorQuantizer_62414464745615
// MI455X (gfx1250) — compile-verified
//
#include <hip/hip_runtime.h>
#include <hip/hip_bf16.h>

// ---------------------------------------------------------------------------
// VectorQuantizer forward for MI455X (gfx1250, wave32, WMMA + async-LDS)
//   inputs : z_[32,256,32,32] f32, conv_w[64,256], conv_b[64], emb[1024,64]
//   outputs: z_q_st[2097152] | loss[1] | sampled[32768] | min_idx[32768] (f32)
// ---------------------------------------------------------------------------

typedef __attribute__((ext_vector_type(2))) float v2f;
typedef __attribute__((ext_vector_type(8))) float v8f;

#define NB    32
#define CIN   256
#define COUT  64
#define HW    1024          // 32*32
#define NE    1024          // codebook size
#define ED    64            // code dim
#define NQ    32768         // B*COUT*HW/ED quantized rows
#define ZELEMS 2097152      // NQ*ED
#define OUT_LOSS    (ZELEMS)
#define OUT_SAMPLED (ZELEMS + 1)
#define OUT_MINIDX  (ZELEMS + 1 + NQ)
#define WPITCH 260          // 256 + 4 : LDS padding (260%64==4 -> conflict-free)
#define PITCH  1032         // 1024 + 8 : LDS padding (1032%64==8 -> conflict-free)

// --- CDNA5 async global->LDS copies (ASYNCcnt), per CDNA5 ISA.
// VDST is a per-lane byte offset into the wave's LDS; both users below have no
// static __shared__, so dynamic-LDS offsets start at 0.
__device__ __forceinline__ void async_load_lds_b128(unsigned ldsOff, const void* g)
{
    asm volatile("global_load_async_to_lds_b128 %0, %1, off"
                 :: "v"(ldsOff), "v"(g) : "memory");
}
__device__ __forceinline__ void async_load_lds_b32(unsigned ldsOff, const void* g)
{
    asm volatile("global_load_async_to_lds_b32 %0, %1, off"
                 :: "v"(ldsOff), "v"(g) : "memory");
}
__device__ __forceinline__ void wait_asynccnt0()
{
    asm volatile("s_wait_asynccnt 0x0" ::: "memory");
}

// ---------------------------------------------------------------------------
// Kernel 1: 1x1 conv as WMMA GEMM.  Per batch: Z[64,1024] = W[64,256] @ X[256,1024]
// Full weight matrix (64x256, padded pitch -> 65KB dynamic LDS, needs CDNA5's
// 320KB/WGP) async-staged once per block; each wave owns one 16-wide hw tile and
// ALL FOUR 16-row o-tiles, so every B-fragment load feeds 4 WMMAs and the 33.5MB
// activation tensor is streamed from HBM exactly once.
// ---------------------------------------------------------------------------
__global__ __launch_bounds__(256) void vq_conv_wmma(
    const float* __restrict__ zin, const float* __restrict__ w,
    const float* __restrict__ bias, float* __restrict__ zout)
{
    extern __shared__ float wlds[];      // [64][260]
    const int tid = threadIdx.x;
    const int blk = blockIdx.x;          // 256 blocks
    const int b   = blk >> 3;            // batch
    const int hg  = blk & 7;             // hw group of 8 waves

    // async-stage all of W[64,256] -> LDS (row pitch 260; 16B chunks stay aligned)
    for (int q = tid; q < COUT * (CIN / 4); q += 256) {
        const int row = q >> 6;              // 0..63
        const int col = (q & 63) << 2;       // 0..252
        async_load_lds_b128((unsigned)((row * WPITCH + col) * 4),
                            w + (size_t)row * CIN + col);
    }
    wait_asynccnt0();
    __syncthreads();

    const int wv   = tid >> 5;
    const int lane = tid & 31;
    const int half = lane >> 4;          // 0: K+0/1, rows r    1: K+2/3, rows 8+r
    const int lm   = lane & 15;
    const int hw0  = (hg * 8 + wv) * 16;
    const float* xb = zin + (size_t)b * CIN * HW;

    v8f acc0 = {0.f,0.f,0.f,0.f,0.f,0.f,0.f,0.f};
    v8f acc1 = acc0, acc2 = acc0, acc3 = acc0;

    #pragma unroll 4
    for (int k0 = 0; k0 < CIN; k0 += 4) {
        const int ka = k0 + half * 2;
        v2f bf;
        bf.x = xb[(size_t)ka * HW + hw0 + lm];
        bf.y = xb[(size_t)(ka + 1) * HW + hw0 + lm];
        // prefetch the B panel ~4 k-steps ahead (4KB stride defeats HW prefetch)
        __builtin_prefetch(&xb[(size_t)(ka + 16) * HW + hw0 + lm], 0, 1);
        const float* arow = &wlds[lm * WPITCH + ka];
        v2f a;
        a.x = arow[0];              a.y = arow[1];
        acc0 = __builtin_amdgcn_wmma_f32_16x16x4_f32(false, a, false, bf, (short)0, acc0, false, false);
        a.x = arow[16 * WPITCH];    a.y = arow[16 * WPITCH + 1];
        acc1 = __builtin_amdgcn_wmma_f32_16x16x4_f32(false, a, false, bf, (short)0, acc1, false, false);
        a.x = arow[32 * WPITCH];    a.y = arow[32 * WPITCH + 1];
        acc2 = __builtin_amdgcn_wmma_f32_16x16x4_f32(false, a, false, bf, (short)0, acc2, false, false);
        a.x = arow[48 * WPITCH];    a.y = arow[48 * WPITCH + 1];
        acc3 = __builtin_amdgcn_wmma_f32_16x16x4_f32(false, a, false, bf, (short)0, acc3, false, false);
    }

    float* zo = zout + (size_t)b * COUT * HW + hw0 + lm;
    #pragma unroll
    for (int r = 0; r < 8; ++r) {
        const int o = r + half * 8;
        zo[(size_t)(o +  0) * HW] = acc0[r] + bias[o +  0];
        zo[(size_t)(o + 16) * HW] = acc1[r] + bias[o + 16];
        zo[(size_t)(o + 32) * HW] = acc2[r] + bias[o + 32];
        zo[(size_t)(o + 48) * HW] = acc3[r] + bias[o + 48];
    }
}

// ---------------------------------------------------------------------------
// Kernel 2: fused distance GEMM (32768x1024x64) + argmin.
// Whole codebook async-transpose-staged into 258 KB of dynamic LDS (CDNA5:
// 320KB/WGP): global reads are lane-consecutive b32s, the per-lane LDS dest
// address performs the [j][k]->[k][j] scatter directly in the async engine.
// Pitch 1032 -> conflict-free B-fragment reads for both 16-lane halves.
// argmin_j d(n,j)  ==  argmax_j (zf_n . e_j - 0.5*||e_j||^2)
// ---------------------------------------------------------------------------
__global__ __launch_bounds__(256) void vq_argmin_wmma(
    const float* __restrict__ zf, const float* __restrict__ emb,
    int* __restrict__ idxout)
{
    extern __shared__ float lds[];       // embT[64][1032] ++ halfnorm[1024]
    float* embT = lds;
    float* hn   = lds + 64 * PITCH;
    const int tid = threadIdx.x;

    // async transpose-stage emb -> embT[k][j]
    for (int q = tid; q < NE * ED; q += 256) {
        const int j = q >> 6;
        const int k = q & 63;
        async_load_lds_b32((unsigned)((k * PITCH + j) * 4), emb + q);
    }
    wait_asynccnt0();
    __syncthreads();
    for (int j = tid; j < NE; j += 256) {
        float s = 0.f;
        #pragma unroll
        for (int k = 0; k < ED; ++k) { float e = embT[k * PITCH + j]; s += e * e; }
        hn[j] = 0.5f * s;
    }
    __syncthreads();

    const int wv = tid >> 5, lane = tid & 31, half = lane >> 4, lm = lane & 15;
    const int m0 = blockIdx.x * 128 + wv * 16;   // 256 blocks x 8 waves x 16 rows

    // preload A fragments: 16 K-steps of this wave's 16 zf rows
    v2f afrag[16];
    const float* zrow = zf + (size_t)(m0 + lm) * ED + half * 2;
    #pragma unroll
    for (int s = 0; s < 16; ++s) { afrag[s].x = zrow[4 * s]; afrag[s].y = zrow[4 * s + 1]; }

    float best[8]; int bidx[8];
    #pragma unroll
    for (int r = 0; r < 8; ++r) { best[r] = -3.402823466e38f; bidx[r] = 0; }

    for (int jt = 0; jt < 64; ++jt) {            // all 1024 codes, 16 at a time
        const int j0 = jt * 16;
        v8f acc = {0.f, 0.f, 0.f, 0.f, 0.f, 0.f, 0.f, 0.f};
        #pragma unroll
        for (int s = 0; s < 16; ++s) {
            const int kb = 4 * s + half * 2;
            v2f bf;
            bf.x = embT[kb * PITCH + j0 + lm];
            bf.y = embT[(kb + 1) * PITCH + j0 + lm];
            acc = __builtin_amdgcn_wmma_f32_16x16x4_f32(false, afrag[s], false, bf,
                                                        (short)0, acc, false, false);
        }
        const int j = j0 + lm;
        const float h = hn[j];
        #pragma unroll
        for (int r = 0; r < 8; ++r) {
            const float sc = acc[r] - h;
            if (sc > best[r]) { best[r] = sc; bidx[r] = j; }   // ascending j: first-min ties
        }
    }

    // butterfly argmax over the 16 lanes sharing a row (wave32, xor<16 stays in half)
    #pragma unroll
    for (int r = 0; r < 8; ++r) {
        #pragma unroll
        for (int off = 1; off < 16; off <<= 1) {
            const float os = __shfl_xor(best[r], off, 32);
            const int   oi = __shfl_xor(bidx[r], off, 32);
            if (os > best[r] || (os == best[r] && oi < bidx[r])) { best[r] = os; bidx[r] = oi; }
        }
    }
    if (lm == 0) {
        #pragma unroll
        for (int r = 0; r < 8; ++r) idxout[m0 + half * 8 + r] = bidx[r];
    }
}

// ---------------------------------------------------------------------------
// Kernel 3: zero the loss scalar and sampled region (harness poisons d_out).
// ---------------------------------------------------------------------------
__global__ void vq_init_out(float* __restrict__ out)
{
    const int i = blockIdx.x * blockDim.x + threadIdx.x;
    if (i < 1 + NE * NB) out[OUT_LOSS + i] = 0.f;   // loss + sampled(32x1024)
}

// ---------------------------------------------------------------------------
// Kernel 4: z_q gather (== z_q_st in fwd) + 1.25*MSE(z_q,z) partial reduction.
// Flat float4 over the 2,097,152 elements: fully coalesced on z / out.
// ---------------------------------------------------------------------------
__global__ __launch_bounds__(256) void vq_gather_loss(
    const float* __restrict__ z, const float* __restrict__ emb,
    const int* __restrict__ idx, float* __restrict__ out)
{
    const int tid = blockIdx.x * 256 + threadIdx.x;      // 2048 blocks
    const size_t e0 = (size_t)tid * 4;
    const int n = (int)(e0 >> 6);
    const int k = (int)(e0 & 63);
    const int j = idx[n];
    const float4 ev = *(const float4*)(emb + (size_t)j * ED + k);
    const float4 zv = *(const float4*)(z + e0);
    *(float4*)(out + e0) = ev;                            // z_q_st = z_q
    const float dx = ev.x - zv.x, dy = ev.y - zv.y, dz = ev.z - zv.z, dw = ev.w - zv.w;
    float sq = dx * dx + dy * dy + dz * dz + dw * dw;

    __shared__ float red[256];
    red[threadIdx.x] = sq;
    __syncthreads();
    for (int s = 128; s > 0; s >>= 1) {
        if (threadIdx.x < s) red[threadIdx.x] += red[threadIdx.x + s];
        __syncthreads();
    }
    if (threadIdx.x == 0)
        atomicAdd(out + OUT_LOSS, red[0] * (1.25f / (float)ZELEMS));
}

// ---------------------------------------------------------------------------
// Kernel 5: sampled scatter (all writes are 1.0 -> race-free) + min_idx as f32.
// ---------------------------------------------------------------------------
__global__ void vq_scatter_idx(const int* __restrict__ idx, float* __restrict__ out)
{
    const int n = blockIdx.x * 256 + threadIdx.x;
    if (n < NQ) {
        const int j = idx[n];
        out[OUT_SAMPLED + j] = 1.0f;          // only flat[0:1024] ever hit (ref quirk)
        out[OUT_MINIDX + n]  = (float)j;
    }
}

// ---------------------------------------------------------------------------
// Kernel 6: contrastive = mean(cosine(emb,emb)) = ||sum_i e_i/||e_i|| ||^2 / N^2
// (exact algebraic collapse of the 1024x1024 cosine matrix).  Single block.
// ---------------------------------------------------------------------------
__global__ __launch_bounds__(256) void vq_contrastive(
    const float* __restrict__ emb, float* __restrict__ out)
{
    __shared__ float s[ED];
    const int tid = threadIdx.x;
    if (tid < ED) s[tid] = 0.f;
    __syncthreads();

    float loc[ED];
    #pragma unroll
    for (int k = 0; k < ED; ++k) loc[k] = 0.f;
    for (int j = tid; j < NE; j += 256) {
        const float* er = emb + (size_t)j * ED;
        float n2 = 0.f;
        #pragma unroll
        for (int k = 0; k < ED; ++k) n2 += er[k] * er[k];
        const float inv = 1.0f / sqrtf(n2);
        #pragma unroll
        for (int k = 0; k < ED; ++k) loc[k] += er[k] * inv;
    }
    #pragma unroll
    for (int k = 0; k < ED; ++k) atomicAdd(&s[k], loc[k]);   // ds_add_f32
    __syncthreads();
    if (tid == 0) {
        float dot = 0.f;
        #pragma unroll
        for (int k = 0; k < ED; ++k) dot += s[k] * s[k];
        atomicAdd(out + OUT_LOSS, dot * (1.0f / ((float)NE * (float)NE)));
    }
}

// ---------------------------------------------------------------------------
extern "C" void kernel_launch(void* const* d_in, const int* in_sizes, int n_in,
                              void* d_out, int out_size, void* d_ws, size_t ws_size,
                              hipStream_t stream)
{
    const float* z_in   = (const float*)d_in[0];
    const float* conv_w = (const float*)d_in[1];
    const float* conv_b = (const float*)d_in[2];
    const float* emb    = (const float*)d_in[3];
    float* out = (float*)d_out;

    float* z  = (float*)d_ws;                                   // 8 MB
    int*   ix = (int*)((char*)d_ws + (size_t)ZELEMS * 4);       // 128 KB

    vq_init_out<<<129, 256, 0, stream>>>(out);
    vq_conv_wmma<<<NB * 8, 256, COUT * WPITCH * sizeof(float), stream>>>(z_in, conv_w, conv_b, z);
    vq_argmin_wmma<<<NQ / 128, 256, (64 * PITCH + NE) * sizeof(float), stream>>>(z, emb, ix);
    vq_gather_loss<<<ZELEMS / (4 * 256), 256, 0, stream>>>(z, emb, ix, out);
    vq_scatter_idx<<<NQ / 256, 256, 0, stream>>>(ix, out);
    vq_contrastive<<<1, 256, 0, stream>>>(emb, out);
}